// VN_DGCNNPredHead_6064493822478
// MI455X (gfx1250) — compile-verified
//
#include <hip/hip_runtime.h>
#include <math.h>

// ---------------------------------------------------------------------------
// VN-DGCNN prediction head for MI455X (gfx1250), wave32.
//  - kNN Gram matrix via V_WMMA_F32_16X16X4_F32, A-slab staged to LDS with the
//    Tensor Data Mover (tensor_load_to_lds + s_wait_tensorcnt), fused top-k.
//  - Factorized VN blocks (PJ/PC trick), VN-BN stats via LDS+global atomics.
//  - conv1/conv2 as WMMA f32 GEMMs, conv3 as dot product.
// ---------------------------------------------------------------------------

typedef __attribute__((ext_vector_type(2))) float v2f;
typedef __attribute__((ext_vector_type(8))) float v8f;
typedef __attribute__((ext_vector_type(4))) unsigned int u32x4;
typedef __attribute__((ext_vector_type(8))) int i32x8;
typedef __attribute__((ext_vector_type(4))) int i32x4;

#if defined(__has_builtin)
#if __has_builtin(__builtin_amdgcn_tensor_load_to_lds) && __has_builtin(__builtin_amdgcn_s_wait_tensorcnt)
#define USE_TDM 1
#endif
#endif

#define BSZ   4
#define CDIM  384
#define C3    128
#define NPTS  2048
#define KNN   20
#define OC1   21
#define NEGS  0.2f
#define EPSV  1e-6f
#define BNEPS 1e-5f
#define NCOLS (BSZ * NPTS)   // 8192 columns for the conv GEMMs

// ---------------- 1: per-point squared norm ----------------
__global__ void xx_kernel(const float* __restrict__ x, float* __restrict__ xx) {
  int i = blockIdx.x * blockDim.x + threadIdx.x;
  if (i >= BSZ * NPTS) return;
  int b = i >> 11, n = i & (NPTS - 1);
  const float* X = x + (size_t)b * CDIM * NPTS + n;
  float s = 0.f;
  for (int c = 0; c < CDIM; ++c) { float v = X[(size_t)c * NPTS]; s += v * v; }
  xx[i] = s;
}

// ---------------- 2: WMMA Gram + fused top-k ----------------
// grid (NPTS/16, BSZ), block 256 (8 waves).
// dyn LDS: negd[16][2048] (128 KB) + aslab[384][16] (24 KB) = 152 KB.
__global__ void gram_topk_kernel(const float* __restrict__ x,
                                 const float* __restrict__ xx,
                                 int* __restrict__ idxOut) {
  extern __shared__ float negd[];            // [16][2048] neg squared distance
  float* aslab = negd + 16 * NPTS;           // [384][16] A slab: aslab[c*16+r]
  const int b       = blockIdx.y;
  const int rowBase = blockIdx.x * 16;
  const int tid  = threadIdx.x;
  const int lane = tid & 31;
  const int wv   = tid >> 5;
  const float* X   = x  + (size_t)b * CDIM * NPTS;
  const float* xxb = xx + (size_t)b * NPTS;
  const int mrow  = lane & 15;
  const int khalf = (lane >> 4) << 1;        // lanes 16-31 hold K+2,K+3

#ifdef USE_TDM
  // Stage A slab (x[c][rowBase..rowBase+16), c=0..383) into LDS via the
  // Tensor Data Mover. One wave issues; TENSORcnt waited before the barrier.
  if (wv == 0) {
    unsigned long long ga = (unsigned long long)(const void*)(X + rowBase);
    unsigned ldsa = (unsigned)(unsigned long long)(const void*)aslab; // LDS byte offset
    u32x4 g0;
    g0[0] = 1u;                                        // count=1, user descriptor
    g0[1] = ldsa;                                      // lds_addr
    g0[2] = (unsigned)(ga & 0xffffffffull);            // global_addr[31:0]
    g0[3] = (unsigned)((ga >> 32) & 0x01ffffffull)     // global_addr[56:32]
            | 0x80000000u;                             // type=2 ("image")
    i32x8 g1;
    g1[0] = 0x00020000;            // data_size=2 -> 4 bytes; no multicast/pad
    g1[1] = (int)(16u << 16);      // tensor_dim0 = 16        (bits 79:48 lo)
    g1[2] = (int)(384u << 16);     // tensor_dim0 hi=0, tensor_dim1 = 384
    g1[3] = (int)(16u << 16);      // tensor_dim1 hi=0, tile_dim0 = 16
    g1[4] = 384;                   // tile_dim1 = 384, tile_dim2 = 0
    g1[5] = 2048;                  // tensor_dim0_stride = 2048 elements
    g1[6] = 0;                     // stride hi / dim1_stride lo
    g1[7] = 0;
    i32x4 gz;
    gz[0] = 0; gz[1] = 0; gz[2] = 0; gz[3] = 0;
    i32x8 gz8;
    gz8[0] = 0; gz8[1] = 0; gz8[2] = 0; gz8[3] = 0;
    gz8[4] = 0; gz8[5] = 0; gz8[6] = 0; gz8[7] = 0;
    // amdgpu-toolchain (clang-23) 6-arg form: (g0, g1, g2, g3, extra, cpol)
    __builtin_amdgcn_tensor_load_to_lds(g0, g1, gz, gz, gz8, 0);
    __builtin_amdgcn_s_wait_tensorcnt(0);
  }
#else
  // Fallback: cooperative copy of the A slab.
  for (int t = tid; t < CDIM * 16; t += 256) {
    int c = t >> 4, r = t & 15;
    aslab[c * 16 + r] = X[(size_t)c * NPTS + rowBase + r];
  }
#endif
  __syncthreads();

  for (int ct = wv; ct < NPTS / 16; ct += 8) {
    const int colBase = ct * 16;
    v8f acc = {};
    __builtin_prefetch(&X[colBase], 0, 0);   // global_prefetch_b8
    for (int k0 = 0; k0 < CDIM; k0 += 4) {
      const float* pb = X + (size_t)(k0 + khalf) * NPTS;
      v2f a, bm;
      a.x  = aslab[(k0 + khalf) * 16 + mrow];        // A from LDS (TDM-staged)
      a.y  = aslab[(k0 + khalf + 1) * 16 + mrow];
      bm.x = pb[colBase + mrow];
      bm.y = pb[NPTS + colBase + mrow];
      acc = __builtin_amdgcn_wmma_f32_16x16x4_f32(
          false, a, false, bm, (short)0, acc, false, false);
    }
    const int halfsel = (lane >> 4) << 3;    // lanes 16-31 hold rows M+8
    const int col = lane & 15;
    const float xxc = xxb[colBase + col];
#pragma unroll
    for (int q = 0; q < 8; ++q) {
      int r = q + halfsel;
      // neg_dist = -(xx_n - 2*G + xx_m) = 2*G - xx_n - xx_m
      negd[r * NPTS + colBase + col] = 2.0f * acc[q] - xxb[rowBase + r] - xxc;
    }
  }
  __syncthreads();

  // top-20 per row (descending, stable = smallest index wins ties)
  for (int r = wv; r < 16; r += 8) {         // each wave: rows wv and wv+8
    float* rowp = negd + r * NPTS;
    int* outp = idxOut + (((size_t)b * NPTS) + rowBase + r) * KNN;
    for (int it = 0; it < KNN; ++it) {
      float best = -3.0e38f; int bi = 0x7fffffff;
      for (int m = lane; m < NPTS; m += 32) {
        float v = rowp[m];
        if (v > best || (v == best && m < bi)) { best = v; bi = m; }
      }
#pragma unroll
      for (int off = 16; off > 0; off >>= 1) {
        float ov = __shfl_xor(best, off, 32);
        int   oi = __shfl_xor(bi, off, 32);
        if (ov > best || (ov == best && oi < bi)) { best = ov; bi = oi; }
      }
      if (lane == 0) { outp[it] = bi; rowp[bi] = -3.0e38f; }
      __syncthreads();
    }
  }
}

// ---------------- 3: factorized VN linear (block 1) ----------------
// p[b,o,v,n,k] = PJ[b,o,v,idx] + PC[b,o,v,n];  PJ = Wa*x3, PC = (Wb-Wa)*x3
__global__ void vn1_lin_kernel(const float* __restrict__ x,
                               const float* __restrict__ wf,
                               const float* __restrict__ wd,
                               float* __restrict__ PJf, float* __restrict__ PCf,
                               float* __restrict__ PJd, float* __restrict__ PCd) {
  size_t i = (size_t)blockIdx.x * blockDim.x + threadIdx.x;
  const size_t total = (size_t)BSZ * OC1 * 3 * NPTS;
  if (i >= total) return;
  int n = (int)(i % NPTS); size_t t = i / NPTS;
  int v = (int)(t % 3); t /= 3;
  int o = (int)(t % OC1); int b = (int)(t / OC1);
  const float* X = x + ((size_t)b * CDIM + v) * NPTS + n;   // x[b][3c+v][n]
  float sjf = 0.f, scf = 0.f, sjd = 0.f, scd = 0.f;
  for (int c = 0; c < C3; ++c) {
    float xv  = X[(size_t)(3 * c) * NPTS];
    float wfa = wf[o * 256 + c], wfb = wf[o * 256 + C3 + c];
    float wda = wd[o * 256 + c], wdb = wd[o * 256 + C3 + c];
    sjf += wfa * xv; scf += (wfb - wfa) * xv;
    sjd += wda * xv; scd += (wdb - wda) * xv;
  }
  PJf[i] = sjf; PCf[i] = scf; PJd[i] = sjd; PCd[i] = scd;
}

// ---------------- stats helpers ----------------
__global__ void zero_kernel(float* p, int n) {
  int i = blockIdx.x * blockDim.x + threadIdx.x;
  if (i < n) p[i] = 0.f;
}

__global__ void finalize_stats_kernel(const float* __restrict__ s1,
                                      const float* __restrict__ s2,
                                      float* __restrict__ mean,
                                      float* __restrict__ invstd, float cnt) {
  int o = threadIdx.x;
  if (o >= OC1) return;
  float m = s1[o] / cnt;
  float var = s2[o] / cnt - m * m;
  mean[o] = m;
  invstd[o] = rsqrtf(var + BNEPS);
}

// ---------------- 4: VN-BN stats for block 1 ----------------
__global__ void vn1_stats_kernel(const int* __restrict__ idx,
                                 const float* __restrict__ PJf,
                                 const float* __restrict__ PCf,
                                 float* __restrict__ s1, float* __restrict__ s2) {
  __shared__ float ls1[OC1], ls2[OC1];
  int tid = threadIdx.x;
  if (tid < OC1) { ls1[tid] = 0.f; ls2[tid] = 0.f; }
  __syncthreads();
  size_t i = (size_t)blockIdx.x * blockDim.x + tid;
  const size_t total = (size_t)BSZ * NPTS * KNN;
  if (i < total) {
    size_t t = i / KNN;
    int n = (int)(t % NPTS); int b = (int)(t / NPTS);
    int j = idx[i];
    for (int o = 0; o < OC1; ++o) {
      float pp = 0.f;
#pragma unroll
      for (int v = 0; v < 3; ++v) {
        size_t base = (((size_t)b * OC1 + o) * 3 + v) * NPTS;
        float p = PJf[base + j] + PCf[base + n];
        pp += p * p;
      }
      float nrm = sqrtf(pp) + EPSV;
      atomicAdd(&ls1[o], nrm);
      atomicAdd(&ls2[o], nrm * nrm);
    }
  }
  __syncthreads();
  if (tid < OC1) { atomicAdd(&s1[tid], ls1[tid]); atomicAdd(&s2[tid], ls2[tid]); }
}

// ---------------- 5: apply VN-BN + VN-LeakyReLU (block 1) ----------------
__global__ void vn1_apply_kernel(const int* __restrict__ idx,
                                 const float* __restrict__ PJf,
                                 const float* __restrict__ PCf,
                                 const float* __restrict__ PJd,
                                 const float* __restrict__ PCd,
                                 const float* __restrict__ mean,
                                 const float* __restrict__ invstd,
                                 float* __restrict__ h1) {
  size_t i = (size_t)blockIdx.x * blockDim.x + threadIdx.x;
  const size_t total = (size_t)BSZ * NPTS * KNN;
  if (i >= total) return;
  size_t t = i / KNN;
  int n = (int)(t % NPTS); int b = (int)(t / NPTS);
  int j = idx[i];
  float* out = h1 + i * 63;
  for (int o = 0; o < OC1; ++o) {
    float p[3], d[3], pp = 0.f, dd = 0.f;
#pragma unroll
    for (int v = 0; v < 3; ++v) {
      size_t base = (((size_t)b * OC1 + o) * 3 + v) * NPTS;
      p[v] = PJf[base + j] + PCf[base + n];
      d[v] = PJd[base + j] + PCd[base + n];
      pp += p[v] * p[v]; dd += d[v] * d[v];
    }
    float nrm = sqrtf(pp) + EPSV;
    float scale = (nrm - mean[o]) * invstd[o] / nrm;
    float dot = 0.f;
#pragma unroll
    for (int v = 0; v < 3; ++v) { p[v] *= scale; dot += p[v] * d[v]; }
    float coef = dot / (dd + EPSV);
#pragma unroll
    for (int v = 0; v < 3; ++v) {
      float proj = p[v] - coef * d[v];
      float sel = (dot >= 0.f) ? p[v] : proj;
      out[o * 3 + v] = NEGS * p[v] + (1.0f - NEGS) * sel;
    }
  }
}

// ---------------- 6: block 2 linear (p2) + stats ----------------
__global__ void vn2_p_stats_kernel(const float* __restrict__ h1,
                                   const float* __restrict__ w2f,
                                   float* __restrict__ p2,
                                   float* __restrict__ s1, float* __restrict__ s2) {
  __shared__ float ls1[OC1], ls2[OC1];
  int tid = threadIdx.x;
  if (tid < OC1) { ls1[tid] = 0.f; ls2[tid] = 0.f; }
  __syncthreads();
  size_t i = (size_t)blockIdx.x * blockDim.x + tid;
  const size_t total = (size_t)BSZ * NPTS * KNN;
  if (i < total) {
    const float* hin = h1 + i * 63;
    float hl[63];
#pragma unroll
    for (int c = 0; c < 63; ++c) hl[c] = hin[c];
    float* outp = p2 + i * 63;
    for (int o = 0; o < OC1; ++o) {
      float pv[3] = {0.f, 0.f, 0.f};
      for (int c = 0; c < OC1; ++c) {
        float w = w2f[o * OC1 + c];
#pragma unroll
        for (int v = 0; v < 3; ++v) pv[v] += w * hl[c * 3 + v];
      }
      float pp = pv[0]*pv[0] + pv[1]*pv[1] + pv[2]*pv[2];
      float nrm = sqrtf(pp) + EPSV;
      atomicAdd(&ls1[o], nrm);
      atomicAdd(&ls2[o], nrm * nrm);
#pragma unroll
      for (int v = 0; v < 3; ++v) outp[o * 3 + v] = pv[v];
    }
  }
  __syncthreads();
  if (tid < OC1) { atomicAdd(&s1[tid], ls1[tid]); atomicAdd(&s2[tid], ls2[tid]); }
}

// ---------------- 7: apply block 2 (overwrites p2 with h2) ----------------
__global__ void vn2_apply_kernel(const float* __restrict__ h1,
                                 const float* __restrict__ w2d,
                                 float* __restrict__ p2h2,
                                 const float* __restrict__ mean,
                                 const float* __restrict__ invstd) {
  size_t i = (size_t)blockIdx.x * blockDim.x + threadIdx.x;
  const size_t total = (size_t)BSZ * NPTS * KNN;
  if (i >= total) return;
  const float* hin = h1 + i * 63;
  float hl[63];
#pragma unroll
  for (int c = 0; c < 63; ++c) hl[c] = hin[c];
  float* pb = p2h2 + i * 63;
  for (int o = 0; o < OC1; ++o) {
    float d[3] = {0.f, 0.f, 0.f};
    for (int c = 0; c < OC1; ++c) {
      float w = w2d[o * OC1 + c];
#pragma unroll
      for (int v = 0; v < 3; ++v) d[v] += w * hl[c * 3 + v];
    }
    float p[3]; float pp = 0.f, dd = 0.f;
#pragma unroll
    for (int v = 0; v < 3; ++v) {
      p[v] = pb[o * 3 + v];
      pp += p[v] * p[v]; dd += d[v] * d[v];
    }
    float nrm = sqrtf(pp) + EPSV;
    float scale = (nrm - mean[o]) * invstd[o] / nrm;
    float dot = 0.f;
#pragma unroll
    for (int v = 0; v < 3; ++v) { p[v] *= scale; dot += p[v] * d[v]; }
    float coef = dot / (dd + EPSV);
#pragma unroll
    for (int v = 0; v < 3; ++v) {
      float proj = p[v] - coef * d[v];
      float sel = (dot >= 0.f) ? p[v] : proj;
      pb[o * 3 + v] = NEGS * p[v] + (1.0f - NEGS) * sel;
    }
  }
}

// ---------------- 8: VN max-pool over k ----------------
__global__ void vn_pool_kernel(const float* __restrict__ h2,
                               const float* __restrict__ wpool,
                               float* __restrict__ h3) {
  int i = blockIdx.x * blockDim.x + threadIdx.x;
  if (i >= BSZ * NPTS) return;
  int b = i >> 11, n = i & (NPTS - 1);
  float bestdot[OC1], bestv[63];
#pragma unroll
  for (int o = 0; o < OC1; ++o) bestdot[o] = -3.0e38f;
  for (int k = 0; k < KNN; ++k) {
    const float* hp = h2 + (((size_t)(b * NPTS + n)) * KNN + k) * 63;
    float hl[63];
#pragma unroll
    for (int c = 0; c < 63; ++c) hl[c] = hp[c];
    for (int o = 0; o < OC1; ++o) {
      float d[3] = {0.f, 0.f, 0.f};
      for (int c = 0; c < OC1; ++c) {
        float w = wpool[o * OC1 + c];
#pragma unroll
        for (int v = 0; v < 3; ++v) d[v] += w * hl[c * 3 + v];
      }
      float dot = hl[o*3]*d[0] + hl[o*3+1]*d[1] + hl[o*3+2]*d[2];
      if (dot > bestdot[o]) {
        bestdot[o] = dot;
#pragma unroll
        for (int v = 0; v < 3; ++v) bestv[o * 3 + v] = hl[o * 3 + v];
      }
    }
  }
  for (int c = 0; c < 63; ++c)
    h3[((size_t)b * 63 + c) * NPTS + n] = bestv[c];
}

// ---------------- 9/10: WMMA GEMM for conv1/conv2 ----------------
// Y[b][o][n] = sum_c W[o][c] * Xin[b][c][n].  grid (NCOLS/128, M/16), block 256.
__global__ void gemm_wmma_kernel(const float* __restrict__ W,
                                 const float* __restrict__ Xin,
                                 float* __restrict__ Y, int M, int Kd) {
  const int tid = threadIdx.x, lane = tid & 31, wv = tid >> 5;
  const int rowBase = blockIdx.y * 16;
  const int colBase = (blockIdx.x * 8 + wv) * 16;
  const int mrow = lane & 15, khalf = (lane >> 4) << 1;
  const int jB = colBase + mrow;               // column this lane feeds into B
  const int bB = jB >> 11, nB = jB & (NPTS - 1);
  v8f acc = {};
  for (int k0 = 0; k0 < Kd; k0 += 4) {
    int c0 = k0 + khalf, c1 = c0 + 1;
    v2f a, bm;
    a.x  = (c0 < Kd) ? W[(rowBase + mrow) * Kd + c0] : 0.f;
    a.y  = (c1 < Kd) ? W[(rowBase + mrow) * Kd + c1] : 0.f;
    bm.x = (c0 < Kd) ? Xin[((size_t)bB * Kd + c0) * NPTS + nB] : 0.f;
    bm.y = (c1 < Kd) ? Xin[((size_t)bB * Kd + c1) * NPTS + nB] : 0.f;
    acc = __builtin_amdgcn_wmma_f32_16x16x4_f32(
        false, a, false, bm, (short)0, acc, false, false);
  }
  const int halfsel = (lane >> 4) << 3;
  const int col = lane & 15;
  const int j = colBase + col;
  const int b = j >> 11, n = j & (NPTS - 1);
#pragma unroll
  for (int q = 0; q < 8; ++q) {
    int o = rowBase + q + halfsel;
    Y[((size_t)b * M + o) * NPTS + n] = acc[q];
  }
}

// per-channel BN stats over (b, n)
__global__ void chan_stats_kernel(const float* __restrict__ Y, int M,
                                  float* __restrict__ mean,
                                  float* __restrict__ invstd) {
  __shared__ float ss[256], sq[256];
  int o = blockIdx.x, tid = threadIdx.x;
  float s = 0.f, q = 0.f;
  for (int i = tid; i < NCOLS; i += 256) {
    int b = i >> 11, n = i & (NPTS - 1);
    float v = Y[((size_t)b * M + o) * NPTS + n];
    s += v; q += v * v;
  }
  ss[tid] = s; sq[tid] = q;
  __syncthreads();
  for (int st = 128; st > 0; st >>= 1) {
    if (tid < st) { ss[tid] += ss[tid + st]; sq[tid] += sq[tid + st]; }
    __syncthreads();
  }
  if (tid == 0) {
    float m = ss[0] / (float)NCOLS;
    float var = sq[0] / (float)NCOLS - m * m;
    mean[o] = m; invstd[o] = rsqrtf(var + BNEPS);
  }
}

__global__ void bn_lrelu_kernel(float* __restrict__ Y, int M,
                                const float* __restrict__ mean,
                                const float* __restrict__ invstd) {
  size_t i = (size_t)blockIdx.x * blockDim.x + threadIdx.x;
  size_t total = (size_t)BSZ * M * NPTS;
  if (i >= total) return;
  int o = (int)((i / NPTS) % M);
  float v = (Y[i] - mean[o]) * invstd[o];
  Y[i] = (v >= 0.f) ? v : NEGS * v;
}

// ---------------- final conv3 ----------------
__global__ void conv3_kernel(const float* __restrict__ Y2,
                             const float* __restrict__ w3,
                             const float* __restrict__ b3,
                             float* __restrict__ out) {
  int i = blockIdx.x * blockDim.x + threadIdx.x;
  if (i >= BSZ * NPTS) return;
  int b = i >> 11, n = i & (NPTS - 1);
  float s = b3[0];
  for (int c = 0; c < 128; ++c)
    s += w3[c] * Y2[((size_t)b * 128 + c) * NPTS + n];
  out[i] = s;
}

// ---------------------------------------------------------------------------
extern "C" void kernel_launch(void* const* d_in, const int* in_sizes, int n_in,
                              void* d_out, int out_size, void* d_ws, size_t ws_size,
                              hipStream_t stream) {
  const float* x     = (const float*)d_in[0];
  const float* w1f   = (const float*)d_in[1];
  const float* w1d   = (const float*)d_in[2];
  const float* w2f   = (const float*)d_in[3];
  const float* w2d   = (const float*)d_in[4];
  const float* wpool = (const float*)d_in[5];
  const float* wc1   = (const float*)d_in[6];
  const float* wc2   = (const float*)d_in[7];
  const float* wc3   = (const float*)d_in[8];
  const float* bc3   = (const float*)d_in[9];
  float* out = (float*)d_out;

  // workspace layout (bytes, 256-aligned); total ~106 MB
  char* ws = (char*)d_ws;
  size_t off = 0;
  auto alloc = [&](size_t bytes) { char* p = ws + off; off = (off + bytes + 255) & ~(size_t)255; return p; };
  float* xx   = (float*)alloc((size_t)BSZ * NPTS * 4);
  int*   idxb = (int*)  alloc((size_t)BSZ * NPTS * KNN * 4);
  size_t pjsz = (size_t)BSZ * OC1 * 3 * NPTS * 4;
  float* PJf = (float*)alloc(pjsz);
  float* PCf = (float*)alloc(pjsz);
  float* PJd = (float*)alloc(pjsz);
  float* PCd = (float*)alloc(pjsz);
  float* stats = (float*)alloc(256 * 4);  // s1a,s2a,m1,is1, s1b,s2b,m2,is2 (8x21)
  float* s1a = stats, *s2a = stats + 21, *m1 = stats + 42, *is1 = stats + 63;
  float* s1b = stats + 84, *s2b = stats + 105, *m2 = stats + 126, *is2 = stats + 147;
  float* cstats = (float*)alloc(1024 * 4); // conv BN: mean1(256) is1(256) mean2(128) is2(128)
  float* cm1 = cstats, *ci1 = cstats + 256, *cm2 = cstats + 512, *ci2 = cstats + 640;
  size_t hsz = (size_t)BSZ * NPTS * KNN * 63 * 4;   // 41.3 MB each
  float* h1 = (float*)alloc(hsz);
  float* h2 = (float*)alloc(hsz);                    // holds p2 then h2
  float* h3 = (float*)alloc((size_t)BSZ * 63 * NPTS * 4);
  float* y1 = (float*)alloc((size_t)BSZ * 256 * NPTS * 4);
  float* y2 = (float*)alloc((size_t)BSZ * 128 * NPTS * 4);
  (void)ws_size; (void)in_sizes; (void)n_in; (void)out_size;

  const size_t nTot = (size_t)BSZ * NPTS * KNN;      // 163840
  const int bnkBlocks = (int)((nTot + 255) / 256);   // 640

  // 0) zero BN sums
  zero_kernel<<<1, 256, 0, stream>>>(stats, 256);
  // 1) squared norms
  xx_kernel<<<(BSZ * NPTS + 255) / 256, 256, 0, stream>>>(x, xx);
  // 2) Gram + top-k (152 KB dynamic LDS: 16x2048 negdist + 384x16 A slab)
  gram_topk_kernel<<<dim3(NPTS / 16, BSZ), 256,
                     (16 * NPTS + CDIM * 16) * sizeof(float), stream>>>(x, xx, idxb);
  // 3) factorized VN linear
  {
    size_t total = (size_t)BSZ * OC1 * 3 * NPTS;
    vn1_lin_kernel<<<(int)((total + 255) / 256), 256, 0, stream>>>(x, w1f, w1d, PJf, PCf, PJd, PCd);
  }
  // 4) VN-BN stats block 1
  vn1_stats_kernel<<<bnkBlocks, 256, 0, stream>>>(idxb, PJf, PCf, s1a, s2a);
  finalize_stats_kernel<<<1, 32, 0, stream>>>(s1a, s2a, m1, is1, (float)nTot);
  // 5) apply block 1
  vn1_apply_kernel<<<bnkBlocks, 256, 0, stream>>>(idxb, PJf, PCf, PJd, PCd, m1, is1, h1);
  // 6) block 2 linear + stats
  vn2_p_stats_kernel<<<bnkBlocks, 256, 0, stream>>>(h1, w2f, h2, s1b, s2b);
  finalize_stats_kernel<<<1, 32, 0, stream>>>(s1b, s2b, m2, is2, (float)nTot);
  // 7) apply block 2
  vn2_apply_kernel<<<bnkBlocks, 256, 0, stream>>>(h1, w2d, h2, m2, is2);
  // 8) VN max-pool
  vn_pool_kernel<<<(BSZ * NPTS + 255) / 256, 256, 0, stream>>>(h2, wpool, h3);
  // 9) conv1: 256x63 GEMM + BN + lrelu
  gemm_wmma_kernel<<<dim3(NCOLS / 128, 256 / 16), 256, 0, stream>>>(wc1, h3, y1, 256, 63);
  chan_stats_kernel<<<256, 256, 0, stream>>>(y1, 256, cm1, ci1);
  bn_lrelu_kernel<<<(int)(((size_t)BSZ * 256 * NPTS + 255) / 256), 256, 0, stream>>>(y1, 256, cm1, ci1);
  // 10) conv2: 128x256 GEMM + BN + lrelu
  gemm_wmma_kernel<<<dim3(NCOLS / 128, 128 / 16), 256, 0, stream>>>(wc2, y1, y2, 128, 256);
  chan_stats_kernel<<<128, 256, 0, stream>>>(y2, 128, cm2, ci2);
  bn_lrelu_kernel<<<(int)(((size_t)BSZ * 128 * NPTS + 255) / 256), 256, 0, stream>>>(y2, 128, cm2, ci2);
  // 11) conv3 -> out
  conv3_kernel<<<(BSZ * NPTS + 255) / 256, 256, 0, stream>>>(y2, wc3, bc3, out);
}